// PaCMAPLoss_231928234516
// MI455X (gfx1250) — compile-verified
//
#include <hip/hip_runtime.h>

// ---------------------------------------------------------------------------
// PaCMAP loss (phase 2) for MI455X / gfx1250.
//
// Bound analysis: emb table (8 MB) lives in the 192 MB L2; only the 280 MB
// index stream touches HBM => ~12 us floor at 23.3 TB/s. Kernel streams
// indices with NT b128 loads, gathers float2 endpoints from L2, reduces
// per-wave with V_WMMA_F32_16X16X4_F32 (A = ones => column sums) + 4 xor
// shuffles, then block partials -> workspace -> deterministic finalize.
// ---------------------------------------------------------------------------

typedef __attribute__((ext_vector_type(2))) float v2f;
typedef __attribute__((ext_vector_type(8))) float v8f;
typedef __attribute__((ext_vector_type(4))) int   v4i;

#define NBLK 1024   // blocks per loss segment
#define NTHR 256    // 8 wave32 per block

// Wave32 full reduction: place each lane's value into one B-matrix slot,
// multiply by A = ones(16x4): D[m,n] = sum_k B[k,n]. The 32 lane values map
// bijectively into the 64 (k,n) slots (other 32 are zero), so the 16 column
// sums partition the lane values; D VGPR0 holds colsum_{lane&15} in every
// lane. Four xor-shuffles over bits 0..3 then sum the 16 column sums.
// Requires EXEC == all ones (call only from fully converged code).
__device__ __forceinline__ float wave_reduce_wmma(float v) {
  v2f a; a[0] = 1.0f; a[1] = 1.0f;   // A = ones (both A VGPRs)
  v2f b; b[0] = v;    b[1] = 0.0f;   // lane value in one B slot
  v8f c = {};
  c = __builtin_amdgcn_wmma_f32_16x16x4_f32(
      /*neg_a=*/false, a, /*neg_b=*/false, b,
      /*c_mod=*/(short)0, c, /*reuse_a=*/false, /*reuse_b=*/false);
  float s = c[0];                    // colsum_{lane & 15}, duplicated halves
  s += __shfl_xor(s, 1);
  s += __shfl_xor(s, 2);
  s += __shfl_xor(s, 4);
  s += __shfl_xor(s, 8);
  return s;                          // every lane: wave total
}

__device__ __forceinline__ float pair_term(const float2* __restrict__ emb2,
                                           int i0, int i1, float cadd,
                                           bool num_is_d) {
  float2 p = emb2[i0];
  float2 q = emb2[i1];
  float dx = p.x - q.x;
  float dy = p.y - q.y;
  float d = fmaf(dx, dx, fmaf(dy, dy, 1.0f));   // sq dist + 1
  return (num_is_d ? d : 1.0f) / (cadd + d);
}

__global__ void pacmap_partials(const float* __restrict__ emb,
                                const int* __restrict__ near_idx, long near_pairs,
                                const int* __restrict__ mid_idx,  long mid_pairs,
                                const int* __restrict__ far_idx,  long far_pairs,
                                float* __restrict__ partials) {
  const int seg = blockIdx.x / NBLK;           // 0=near, 1=mid, 2=far
  const int blk = blockIdx.x - seg * NBLK;

  const int* idx; long pairs; float cadd; bool num_is_d;
  if (seg == 0)      { idx = near_idx; pairs = near_pairs; cadd = 10.0f;    num_is_d = true;  }
  else if (seg == 1) { idx = mid_idx;  pairs = mid_pairs;  cadd = 10000.0f; num_is_d = true;  }
  else               { idx = far_idx;  pairs = far_pairs;  cadd = 1.0f;     num_is_d = false; }

  const float2* emb2 = (const float2*)emb;
  const long tid    = (long)blk * NTHR + threadIdx.x;
  const long stride = (long)NBLK * NTHR;

  float acc = 0.0f;

  // Main loop: 2 pairs per iteration via NT 128-bit index loads
  // (keeps the 280 MB index stream out of the L2 working set).
  const long quads = pairs >> 1;
  const v4i* idx4 = (const v4i*)idx;
  for (long i = tid; i < quads; i += stride) {
    v4i qd = __builtin_nontemporal_load(&idx4[i]);
    acc += pair_term(emb2, qd[0], qd[1], cadd, num_is_d);
    acc += pair_term(emb2, qd[2], qd[3], cadd, num_is_d);
  }
  // Tail (odd pair count)
  const int2* idx2 = (const int2*)idx;
  for (long i = (quads << 1) + tid; i < pairs; i += stride) {
    int2 p = idx2[i];
    acc += pair_term(emb2, p.x, p.y, cadd, num_is_d);
  }

  // Fully converged here: EXEC = all ones for WMMA.
  float wsum = wave_reduce_wmma(acc);

  __shared__ float lds[NTHR / 32];
  const int lane = threadIdx.x & 31;
  const int wid  = threadIdx.x >> 5;
  if (lane == 0) lds[wid] = wsum;
  __syncthreads();
  if (threadIdx.x == 0) {
    float s = 0.0f;
    #pragma unroll
    for (int w = 0; w < NTHR / 32; ++w) s += lds[w];
    partials[blockIdx.x] = s;        // deterministic: no atomics
  }
}

__global__ void pacmap_finalize(const float* __restrict__ partials,
                                float* __restrict__ out) {
  __shared__ float lds[NTHR / 32];
  const float wgt[3] = {3.0f, 3.0f, 1.0f};   // neighbor, midnear, far
  for (int s = 0; s < 3; ++s) {
    float acc = 0.0f;
    for (int i = threadIdx.x; i < NBLK; i += NTHR)   // NBLK % NTHR == 0: uniform
      acc += partials[s * NBLK + i];
    float wsum = wave_reduce_wmma(acc);              // EXEC all ones
    const int lane = threadIdx.x & 31;
    const int wid  = threadIdx.x >> 5;
    if (lane == 0) lds[wid] = wsum;
    __syncthreads();
    if (threadIdx.x == 0) {
      float t = 0.0f;
      #pragma unroll
      for (int w = 0; w < NTHR / 32; ++w) t += lds[w];
      out[s] = t * wgt[s];
    }
    __syncthreads();
  }
}

extern "C" void kernel_launch(void* const* d_in, const int* in_sizes, int n_in,
                              void* d_out, int out_size, void* d_ws, size_t ws_size,
                              hipStream_t stream) {
  const float* emb      = (const float*)d_in[0];
  const int*   near_idx = (const int*)d_in[1];
  const int*   mid_idx  = (const int*)d_in[2];
  const int*   far_idx  = (const int*)d_in[3];
  const long near_pairs = (long)in_sizes[1] / 2;
  const long mid_pairs  = (long)in_sizes[2] / 2;
  const long far_pairs  = (long)in_sizes[3] / 2;

  float* partials = (float*)d_ws;    // 3 * NBLK floats = 12 KB scratch

  pacmap_partials<<<3 * NBLK, NTHR, 0, stream>>>(
      emb, near_idx, near_pairs, mid_idx, mid_pairs, far_idx, far_pairs,
      partials);
  pacmap_finalize<<<1, NTHR, 0, stream>>>(partials, (float*)d_out);
}